// OSIM_70351564308841
// MI455X (gfx1250) — compile-verified
//
#include <hip/hip_runtime.h>
#include <cstdint>
#include <utility>

#define NJ   22
#define NC   33
#define TPB  128
#define NOUT 195   // (22 body + 43 joint) * 3

static constexpr int PARENT[NJ] = {-1,0,0,0,1,1,3,2,2,3,5,6,8,9,10,11,12,13,14,15,16,17};

// ---------------- CDNA5 async global<->LDS helpers (ASYNCcnt path) -----------
__device__ __forceinline__ uint32_t lds_byte_addr(const void* p) {
  // Flat shared-aperture addresses carry the LDS byte offset in addr[31:0]
  return (uint32_t)(uintptr_t)p;
}

template <int OFF>
__device__ __forceinline__ void async_g2l_b128(uint32_t lds_byte, uint64_t gaddr) {
  // INST_OFFSET is applied to BOTH the LDS and the global address (ISA 15.18.3):
  asm volatile("global_load_async_to_lds_b128 %0, %1, off offset:%2"
               :: "v"(lds_byte), "v"(gaddr), "n"(OFF) : "memory");
}

template <int OFF>
__device__ __forceinline__ void async_l2g_b128(uint64_t gaddr, uint32_t lds_byte) {
  asm volatile("global_store_async_from_lds_b128 %0, %1, off offset:%2"
               :: "v"(gaddr), "v"(lds_byte), "n"(OFF) : "memory");
}

__device__ __forceinline__ void wait_async0() {
  asm volatile("s_wait_asynccnt 0" ::: "memory");
}

template <int STRIDE, int... Ks>
__device__ __forceinline__ void g2l_seq(uint32_t lb, uint64_t gb,
                                        std::integer_sequence<int, Ks...>) {
  ((async_g2l_b128<Ks * STRIDE>(lb, gb)), ...);
}

template <int STRIDE, int... Ks>
__device__ __forceinline__ void l2g_seq(uint64_t gb, uint32_t lb,
                                        std::integer_sequence<int, Ks...>) {
  ((async_l2g_b128<Ks * STRIDE>(gb, lb)), ...);
}

// Copy NBYTES (16B-multiple) global -> LDS, whole block cooperating.
// NBYTES = TPB*16*FULL + 16*REM, REM < TPB.
template <int NBYTES>
__device__ __forceinline__ void stage_in(const float* __restrict__ g,
                                         const float* l, int tid) {
  constexpr int NCHUNK = NBYTES >> 4;
  constexpr int FULL   = NCHUNK / TPB;
  constexpr int REM    = NCHUNK % TPB;
  const uint32_t lb = lds_byte_addr(l) + (tid << 4);
  const uint64_t gb = (uint64_t)(uintptr_t)g + ((uint64_t)tid << 4);
  g2l_seq<TPB * 16>(lb, gb, std::make_integer_sequence<int, FULL>{});
  if constexpr (REM > 0) {
    if (tid < REM) async_g2l_b128<FULL * TPB * 16>(lb, gb);
  }
}

// Copy NBYTES (16B-multiple) LDS -> global, whole block cooperating.
template <int NBYTES>
__device__ __forceinline__ void stage_out(float* __restrict__ g,
                                          const float* l, int tid) {
  constexpr int NCHUNK = NBYTES >> 4;
  constexpr int FULL   = NCHUNK / TPB;
  constexpr int REM    = NCHUNK % TPB;
  const uint32_t lb = lds_byte_addr(l) + (tid << 4);
  const uint64_t gb = (uint64_t)(uintptr_t)g + ((uint64_t)tid << 4);
  l2g_seq<TPB * 16>(gb, lb, std::make_integer_sequence<int, FULL>{});
  if constexpr (REM > 0) {
    if (tid < REM) async_l2g_b128<FULL * TPB * 16>(gb, lb);
  }
}

// ---------------- small matrix helpers ---------------------------------------
__device__ __forceinline__ void mm33(float* __restrict__ d,
                                     const float* __restrict__ a,
                                     const float* __restrict__ b) {
#pragma unroll
  for (int r = 0; r < 3; ++r)
#pragma unroll
    for (int c = 0; c < 3; ++c)
      d[r*3+c] = fmaf(a[r*3+0], b[0*3+c], fmaf(a[r*3+1], b[1*3+c], a[r*3+2]*b[2*3+c]));
}

__device__ __forceinline__ void mv33(float* __restrict__ d,
                                     const float* __restrict__ a,
                                     const float* __restrict__ v) {
#pragma unroll
  for (int r = 0; r < 3; ++r)
    d[r] = fmaf(a[r*3+0], v[0], fmaf(a[r*3+1], v[1], a[r*3+2]*v[2]));
}

// R = Rx(ex) @ Ry(ey) @ Rz(ez)
__device__ __forceinline__ void euler_xyz(float* __restrict__ R,
                                          float ex, float ey, float ez) {
  float sx, cx, sy, cy, sz, cz;
  __sincosf(ex, &sx, &cx);
  __sincosf(ey, &sy, &cy);
  __sincosf(ez, &sz, &cz);
  R[0] = cy*cz;                      R[1] = -cy*sz;                     R[2] = sy;
  R[3] = fmaf(sx*sy, cz,  cx*sz);    R[4] = fmaf(-sx*sy, sz, cx*cz);    R[5] = -sx*cy;
  R[6] = fmaf(-cx*sy, cz, sx*sz);    R[7] = fmaf(cx*sy, sz,  sx*cz);    R[8] = cx*cy;
}

// ---------------- kernel ------------------------------------------------------
__global__ void __launch_bounds__(TPB)
fk_cdna5_kernel(const float* __restrict__ coords,    // (B,33)
                const float* __restrict__ scales,    // (B,22,3)
                const float* __restrict__ XBpJp_r,   // (22,3,3)
                const float* __restrict__ XBpJp_t,   // (22,3,1)
                const float* __restrict__ XJcBc_r,   // (22,3,3)
                const float* __restrict__ XBcJc_t,   // (22,3,1)
                const float* __restrict__ XBBcen_t,  // (22,3,1)
                const float* __restrict__ XG_r,      // (3,3)
                const float* __restrict__ XG_t,      // (3,1)
                const float* __restrict__ rot_a,     // (22,3)
                const float* __restrict__ rot_b,     // (22,3)
                const float* __restrict__ trans_a,   // (22,3)
                const float* __restrict__ trans_b,   // (22,3)
                const int*   __restrict__ rot_idx,   // (22,3)
                const int*   __restrict__ trans_idx, // (22,3)
                float* __restrict__ out,             // (B,65,3)
                int Btot) {
  __shared__ __align__(16) float shC[TPB * NC];        //  16,896 B
  __shared__ __align__(16) float shS[TPB * NJ * 3];    //  33,792 B
  __shared__ __align__(16) float shO[TPB * NOUT];      //  99,840 B

  const int tid = threadIdx.x;
  const int b0  = blockIdx.x * TPB;
  const bool full = (b0 + TPB <= Btot);

  // ---- Stage inputs into LDS: block rows are contiguous in global memory ----
  if (full) {
    stage_in<TPB * NC * 4>(coords + (size_t)b0 * NC, shC, tid);
    stage_in<TPB * NJ * 3 * 4>(scales + (size_t)b0 * NJ * 3, shS, tid);
    wait_async0();
  } else if (b0 + tid < Btot) {                                 // rare tail
    for (int k = 0; k < NC; ++k)     shC[tid*NC + k]   = coords[(size_t)(b0+tid)*NC + k];
    for (int k = 0; k < NJ*3; ++k)   shS[tid*NJ*3 + k] = scales[(size_t)(b0+tid)*NJ*3 + k];
  }
  __syncthreads();

  // ---- Per-element FK chain (fully unrolled; constants are wave-uniform) ----
  if (b0 + tid < Btot) {
    const float* Crow = &shC[tid * NC];
    const float* Srow = &shS[tid * NJ * 3];
    float* O = &shO[tid * NOUT];

    float bR[NJ][9];
    float bT[NJ][3];

#pragma unroll
    for (int j = 0; j < NJ; ++j) {
      const int p = PARENT[j];
      float Rp[9], tp[3], sp[3];
      if (p < 0) {
#pragma unroll
        for (int k = 0; k < 9; ++k) Rp[k] = XG_r[k];
#pragma unroll
        for (int k = 0; k < 3; ++k) { tp[k] = XG_t[k]; sp[k] = 1.0f; }
      } else {
#pragma unroll
        for (int k = 0; k < 9; ++k) Rp[k] = bR[p][k];
#pragma unroll
        for (int k = 0; k < 3; ++k) { tp[k] = bT[p][k]; sp[k] = fabsf(Srow[p*3 + k]); }
      }
      float sc[3];
#pragma unroll
      for (int k = 0; k < 3; ++k) sc[k] = fabsf(Srow[j*3 + k]);

      // R_GJp = Rp @ X_BpJp_r[j];  t_GJp = tp + Rp @ (X_BpJp_t[j] * sp)
      float RGJp[9];
      mm33(RGJp, Rp, &XBpJp_r[j*9]);
      float v0[3];
#pragma unroll
      for (int k = 0; k < 3; ++k) v0[k] = XBpJp_t[j*3 + k] * sp[k];
      float tGJp[3];
      mv33(tGJp, Rp, v0);
#pragma unroll
      for (int k = 0; k < 3; ++k) tGJp[k] += tp[k];

      if (p >= 0) {                       // joint_pos entry m = 2j-1
#pragma unroll
        for (int k = 0; k < 3; ++k) O[66 + (2*j - 1)*3 + k] = tGJp[k];
      }

      // joint-local rotation / translation from coords
      float e[3], tl[3];
#pragma unroll
      for (int k = 0; k < 3; ++k) {
        const int ri = rot_idx[j*3 + k];
        const int ti = trans_idx[j*3 + k];
        e[k]  = fmaf(rot_a[j*3 + k],   Crow[ri], rot_b[j*3 + k]);
        tl[k] = fmaf(trans_a[j*3 + k], Crow[ti], trans_b[j*3 + k]) * sp[k];
      }
      float Rloc[9];
      euler_xyz(Rloc, e[0], e[1], e[2]);

      float RGJc[9];
      mm33(RGJc, RGJp, Rloc);
      float tGJc[3];
      mv33(tGJc, RGJp, tl);
#pragma unroll
      for (int k = 0; k < 3; ++k) tGJc[k] += tGJp[k];

      const int mjc = (j == 0) ? 0 : 2*j;  // joint_pos entry for t_GJc
#pragma unroll
      for (int k = 0; k < 3; ++k) O[66 + mjc*3 + k] = tGJc[k];

      // body frame
      float RGBc[9];
      mm33(RGBc, RGJc, &XJcBc_r[j*9]);
      float w[3], dw[3];
#pragma unroll
      for (int k = 0; k < 3; ++k) w[k] = XBcJc_t[j*3 + k] * sc[k];
      mv33(dw, RGBc, w);
      float tGBc[3];
#pragma unroll
      for (int k = 0; k < 3; ++k) tGBc[k] = tGJc[k] - dw[k];

#pragma unroll
      for (int k = 0; k < 9; ++k) bR[j][k] = RGBc[k];
#pragma unroll
      for (int k = 0; k < 3; ++k) bT[j][k] = tGBc[k];

      // body_pos
      float u[3], du[3];
#pragma unroll
      for (int k = 0; k < 3; ++k) u[k] = XBBcen_t[j*3 + k] * sc[k];
      mv33(du, RGBc, u);
#pragma unroll
      for (int k = 0; k < 3; ++k) O[j*3 + k] = tGBc[k] + du[k];
    }
  }
  __syncthreads();

  // ---- Stream block-contiguous output tile LDS -> global (async) -----------
  if (full) {
    stage_out<TPB * NOUT * 4>(out + (size_t)b0 * NOUT, shO, tid);
    wait_async0();
  } else if (b0 + tid < Btot) {
    const float* O = &shO[tid * NOUT];
    for (int k = 0; k < NOUT; ++k) out[(size_t)(b0 + tid) * NOUT + k] = O[k];
  }
}

// ---------------- launcher ----------------------------------------------------
extern "C" void kernel_launch(void* const* d_in, const int* in_sizes, int n_in,
                              void* d_out, int out_size, void* d_ws, size_t ws_size,
                              hipStream_t stream) {
  const float* coords    = (const float*)d_in[0];
  const float* scales    = (const float*)d_in[1];
  const float* XBpJp_r   = (const float*)d_in[2];
  const float* XBpJp_t   = (const float*)d_in[3];
  const float* XJcBc_r   = (const float*)d_in[4];
  const float* XBcJc_t   = (const float*)d_in[5];
  const float* XBBcen_t  = (const float*)d_in[6];
  const float* XG_r      = (const float*)d_in[7];
  const float* XG_t      = (const float*)d_in[8];
  const float* rot_a     = (const float*)d_in[9];
  const float* rot_b     = (const float*)d_in[10];
  const float* trans_a   = (const float*)d_in[11];
  const float* trans_b   = (const float*)d_in[12];
  const int*   rot_idx   = (const int*)d_in[13];
  const int*   trans_idx = (const int*)d_in[14];
  float* out = (float*)d_out;

  const int Btot = in_sizes[0] / NC;
  const int grid = (Btot + TPB - 1) / TPB;

  fk_cdna5_kernel<<<grid, TPB, 0, stream>>>(
      coords, scales, XBpJp_r, XBpJp_t, XJcBc_r, XBcJc_t, XBBcen_t,
      XG_r, XG_t, rot_a, rot_b, trans_a, trans_b, rot_idx, trans_idx,
      out, Btot);
}